// MemRNN_8057358647558
// MI455X (gfx1250) — compile-verified
//
#include <hip/hip_runtime.h>
#include <math.h>

// Problem sizes (fixed by the reference).
#define TT 1024
#define BB 64
#define II 256
#define HH 256
#define BH (BB * HH)        // 16384

#define NBLK 64             // persistent blocks
#define NTHR 1024           // threads per persistent block
#define GTH (NBLK * NTHR)   // 65536 grid threads

typedef __attribute__((ext_vector_type(2))) float v2f;
typedef __attribute__((ext_vector_type(8))) float v8f;

// ---------------------------------------------------------------------------
// Hardware tanh (gfx1250 V_TANH_F32, TRANS pipe). v_nop satisfies the
// 1-independent-op TRANS hazard rule before the result is consumed.
// ---------------------------------------------------------------------------
__device__ __forceinline__ float htanh(float x) {
  float r;
  asm("v_tanh_f32 %0, %1\n\tv_nop" : "=v"(r) : "v"(x));
  return r;
}

// ---------------------------------------------------------------------------
// Software grid barrier: monotonic arrival counter + generation word.
// cnt/gen zeroed by init kernel each launch -> deterministic across calls.
// ---------------------------------------------------------------------------
__device__ __forceinline__ void gsync(unsigned* cnt, unsigned* gen,
                                      unsigned nblocks, unsigned epoch) {
  __syncthreads();
  if (threadIdx.x == 0) {
    __threadfence();  // release my writes to device scope
    unsigned a = atomicAdd(cnt, 1u);  // monotonic, never reset -> race-free
    if (a == epoch * nblocks + (nblocks - 1)) {
      __atomic_store_n(gen, epoch + 1u, __ATOMIC_RELEASE);
    } else {
      while (__atomic_load_n(gen, __ATOMIC_ACQUIRE) <= epoch) {
        __builtin_amdgcn_s_sleep(2);
      }
    }
    __threadfence();  // acquire other blocks' writes
  }
  __syncthreads();
}

__global__ void init_barrier_kernel(unsigned* bar) {
  if (threadIdx.x < 2) bar[threadIdx.x] = 0u;
}

// ---------------------------------------------------------------------------
// fp32 WMMA fragment helpers (V_WMMA_F32_16X16X4_F32).
// A 16x4 (MxK) row-major: lane m = lane&15, K pair = k0 + 2*(lane>>4).
// B 4x16 (KxN) = W^T with W row-major [N,K]: lane n = lane&15, same K pairing.
// C/D 16x16: VGPR r -> row r + 8*(lane>>4), col = lane&15.
// ---------------------------------------------------------------------------
__device__ __forceinline__ v8f wmma_k_loop(const float* __restrict__ ap,
                                           const float* __restrict__ bp,
                                           int ksteps, v8f c) {
  for (int k = 0; k < ksteps; ++k) {
    v2f a = { ap[0], ap[1] };
    v2f b = { bp[0], bp[1] };
    c = __builtin_amdgcn_wmma_f32_16x16x4_f32(false, a, false, b,
                                              (short)0, c, false, false);
    ap += 4; bp += 4;
  }
  return c;
}

// 16x16 tile of C = A[64x256] * W^T computed by one 256-thread sub-block
// (8 waves split K=256 into 32-wide chunks; partials reduced through LDS).
// Returns the element owned by this thread: row m0+t256/16, col n0+t256%16.
__device__ __forceinline__ float subblock_gemm_tile(const float* __restrict__ A, int lda,
                                                    const float* __restrict__ W, int ldw,
                                                    float* red /*LDS 8*256*/,
                                                    int t256, int m0, int n0) {
  const int w    = t256 >> 5;
  const int lane = t256 & 31;
  const int half = lane >> 4;
  const int l15  = lane & 15;

  v8f c = {};
  const float* ap = A + (m0 + l15) * lda + w * 32 + 2 * half;
  const float* bp = W + (n0 + l15) * ldw + w * 32 + 2 * half;
  c = wmma_k_loop(ap, bp, 8, c);  // 8 WMMAs per wave

#pragma unroll
  for (int r = 0; r < 8; ++r)
    red[w * 256 + (r + 8 * half) * 16 + l15] = c[r];
  __syncthreads();

  float s = 0.f;
#pragma unroll
  for (int w2 = 0; w2 < 8; ++w2) s += red[w2 * 256 + t256];
  __syncthreads();  // LDS reusable afterwards
  return s;
}

// ---------------------------------------------------------------------------
// Kernel 1: Ux[t,b,:] = x[t,b,:] @ U_w^T + U_b   (M=T*B=65536, N=H, K=I)
// One 16x16 tile per wave, 64 WMMAs along K. 8 waves/block -> 8192 blocks.
// ---------------------------------------------------------------------------
__global__ void __launch_bounds__(256)
ux_gemm_kernel(const float* __restrict__ x, const float* __restrict__ Uw,
               const float* __restrict__ Ub, float* __restrict__ Ux) {
  const int tid  = threadIdx.x;
  const int lane = tid & 31;
  const int half = lane >> 4;
  const int l15  = lane & 15;

  const int wtile = blockIdx.x * 8 + (tid >> 5);  // 0 .. 65535
  const int m0 = (wtile >> 4) << 4;               // row tile in [0, 65536)
  const int n0 = (wtile & 15) << 4;               // col tile in [0, 256)

  v8f c = {};
  const float* ap = x  + (size_t)(m0 + l15) * II + 2 * half;
  const float* bp = Uw + (size_t)(n0 + l15) * II + 2 * half;
  c = wmma_k_loop(ap, bp, II / 4, c);  // 64 WMMAs

  const float bias = Ub[n0 + l15];
#pragma unroll
  for (int r = 0; r < 8; ++r) {
    int m = m0 + r + 8 * half;
    Ux[(size_t)m * HH + n0 + l15] = c[r] + bias;
  }
}

// ---------------------------------------------------------------------------
// Kernel 2: persistent cooperative recurrence. 64 blocks x 1024 threads.
// mem == d_out ([T,B,H]); uam/q/st/e/ctpart are workspace.
// Each block acts as 4 virtual 256-thread sub-blocks for the step GEMMs
// (blocks 0..15 cover all 64 16x16 tiles of the 64x256 step outputs).
// ---------------------------------------------------------------------------
__global__ void __launch_bounds__(NTHR)
memrnn_persistent_kernel(const float* __restrict__ Ux,
                         const float* __restrict__ Vw,
                         const float* __restrict__ Uaw,
                         const float* __restrict__ Vaw,
                         const float* __restrict__ vvec,
                         float* __restrict__ mem,
                         float* __restrict__ uam,
                         float* __restrict__ q,
                         float* __restrict__ st,
                         float* __restrict__ e,
                         float* __restrict__ ctpart,
                         unsigned* __restrict__ bar) {
  __shared__ float red[4 * 8 * 256];   // 32 KB: 4 sub-blocks x (8 waves x 256)
  __shared__ float sred[NTHR];         // 4 KB reduction scratch

  const int tid  = threadIdx.x;
  const int bid  = blockIdx.x;
  const int gid  = bid * NTHR + tid;
  const int sub  = tid >> 8;           // virtual sub-block 0..3
  const int t256 = tid & 255;

  unsigned* cnt = bar;
  unsigned* gen = bar + 1;
  unsigned  ep  = 0;

  // This thread's GEMM-tile ownership (valid when bid < 16).
  const int tl   = bid * 4 + sub;              // tile 0..63
  const int m0   = (tl >> 4) << 4;             // row tile in [0,64)
  const int n0   = (tl & 15) << 4;             // col tile in [0,256)
  const int rrow = m0 + t256 / 16;             // owned output row (b)
  const int rcol = n0 + (t256 & 15);           // owned output col (h)
  float* redsub  = red + sub * 2048;

  // ---- t = 0: st = Ux[0]; mem[0] = tanh(Ux[0]) ----
  for (int idx = gid; idx < BH; idx += GTH) {
    float u = Ux[idx];
    st[idx]  = u;
    mem[idx] = tanhf(u);                       // precise: feeds d_out
  }
  gsync(cnt, gen, NBLK, ep++);

  // ---- uam[0] = mem[0] @ Ua_w^T ----
  if (bid < 16) {
    float val = subblock_gemm_tile(mem, HH, Uaw, HH, redsub, t256, m0, n0);
    uam[(size_t)rrow * HH + rcol] = val;
  }
  gsync(cnt, gen, NBLK, ep++);

  for (int t = 1; t < TT; ++t) {
    // P1: q = st @ Va_w^T
    if (bid < 16) {
      float val = subblock_gemm_tile(st, HH, Vaw, HH, redsub, t256, m0, n0);
      q[(size_t)rrow * HH + rcol] = val;
    }
    gsync(cnt, gen, NBLK, ep++);

    // P2: scores e[tt,b] = sum_h v[h] * tanh(q[b,h] + uam[tt,b,h]),  tt < t
    {
      const int pairs = t * BB;
      for (int p = gid; p < pairs; p += GTH) {
        const int ts = p >> 6;           // p / B
        const int b  = p & 63;
        const float4* qp = (const float4*)(q + (size_t)b * HH);
        const float4* up = (const float4*)(uam + (size_t)ts * BH + (size_t)b * HH);
        const float4* vp = (const float4*)vvec;
        float acc = 0.f;
        for (int h4 = 0; h4 < HH / 4; ++h4) {
          float4 qv = qp[h4], uv = up[h4], vv = vp[h4];
          acc += vv.x * htanh(qv.x + uv.x);
          acc += vv.y * htanh(qv.y + uv.y);
          acc += vv.z * htanh(qv.z + uv.z);
          acc += vv.w * htanh(qv.w + uv.w);
        }
        e[p] = acc;                      // layout e[tt*B + b]
      }
    }
    gsync(cnt, gen, NBLK, ep++);

    // P3: per-batch softmax over time, normalized in place (block b = bid).
    {
      const int b = bid;                 // NBLK == BB == 64
      float mx = -INFINITY;
      for (int s0 = tid; s0 < t; s0 += NTHR) mx = fmaxf(mx, e[s0 * BB + b]);
      sred[tid] = mx; __syncthreads();
      for (int off = NTHR / 2; off > 0; off >>= 1) {
        if (tid < off) sred[tid] = fmaxf(sred[tid], sred[tid + off]);
        __syncthreads();
      }
      mx = sred[0]; __syncthreads();

      float sm = 0.f;
      for (int s0 = tid; s0 < t; s0 += NTHR) sm += expf(e[s0 * BB + b] - mx);
      sred[tid] = sm; __syncthreads();
      for (int off = NTHR / 2; off > 0; off >>= 1) {
        if (tid < off) sred[tid] += sred[tid + off];
        __syncthreads();
      }
      const float inv = 1.f / sred[0]; __syncthreads();

      for (int s0 = tid; s0 < t; s0 += NTHR)
        e[s0 * BB + b] = expf(e[s0 * BB + b] - mx) * inv;  // alphas
    }
    gsync(cnt, gen, NBLK, ep++);

    // P4a: 4-way time-split partial context sums (deterministic per slot).
    //      slot = g*BH + idx; group g accumulates tt = g, g+4, g+8, ...
    for (int slot = gid; slot < 4 * BH; slot += GTH) {
      const int g   = slot >> 14;        // 0..3
      const int idx = slot & (BH - 1);
      const int b   = idx >> 8;          // H = 256
      const float* mp = mem + idx;       // stride BH over tt (coalesced in idx)
      const float* ap = e + b;           // stride B over tt (broadcast per b)
      float acc = 0.f;
      for (int s0 = g; s0 < t; s0 += 4)
        acc = fmaf(ap[(size_t)s0 * BB], mp[(size_t)s0 * BH], acc);
      ctpart[slot] = acc;
    }
    gsync(cnt, gen, NBLK, ep++);

    // P4b: st = mem[t-1] + sum_g ctpart[g]
    for (int idx = gid; idx < BH; idx += GTH) {
      float c0 = ctpart[idx] + ctpart[BH + idx];
      float c1 = ctpart[2 * BH + idx] + ctpart[3 * BH + idx];
      st[idx] = mem[(size_t)(t - 1) * BH + idx] + (c0 + c1);
    }
    gsync(cnt, gen, NBLK, ep++);

    // P5: h = tanh(Ux[t] + st @ V_w^T);  mem[t] = h
    if (bid < 16) {
      float val = subblock_gemm_tile(st, HH, Vw, HH, redsub, t256, m0, n0);
      const size_t o = (size_t)t * BH + (size_t)rrow * HH + rcol;
      mem[o] = tanhf(Ux[o] + val);       // precise: feeds d_out
    }
    gsync(cnt, gen, NBLK, ep++);

    // P6: uam[t] = mem[t] @ Ua_w^T
    if (bid < 16) {
      float val = subblock_gemm_tile(mem + (size_t)t * BH, HH, Uaw, HH,
                                     redsub, t256, m0, n0);
      uam[(size_t)t * BH + (size_t)rrow * HH + rcol] = val;
    }
    gsync(cnt, gen, NBLK, ep++);
  }
}

// ---------------------------------------------------------------------------
// Launch: init barrier -> Ux GEMM (parallel, WMMA) -> persistent recurrence.
// ---------------------------------------------------------------------------
extern "C" void kernel_launch(void* const* d_in, const int* in_sizes, int n_in,
                              void* d_out, int out_size, void* d_ws, size_t ws_size,
                              hipStream_t stream) {
  const float* x    = (const float*)d_in[0];
  const float* U_w  = (const float*)d_in[1];
  const float* U_b  = (const float*)d_in[2];
  const float* V_w  = (const float*)d_in[3];
  const float* Ua_w = (const float*)d_in[4];
  const float* Va_w = (const float*)d_in[5];
  const float* v    = (const float*)d_in[6];

  float* mem = (float*)d_out;                        // [T,B,H] == output

  float* Ux     = (float*)d_ws;                      // T*B*H
  float* uam    = Ux + (size_t)TT * BH;              // T*B*H
  float* q      = uam + (size_t)TT * BH;             // B*H
  float* st     = q + BH;                            // B*H
  float* e      = st + BH;                           // T*B
  float* ctpart = e + (size_t)TT * BB;               // 4*B*H
  unsigned* bar = (unsigned*)(ctpart + 4 * BH);      // 2 uints

  (void)in_sizes; (void)n_in; (void)out_size; (void)ws_size;

  init_barrier_kernel<<<1, 64, 0, stream>>>(bar);

  // M = T*B = 65536 rows -> 4096*16 wave tiles / 8 waves = 8192 blocks
  ux_gemm_kernel<<<8192, 256, 0, stream>>>(x, U_w, U_b, Ux);

  memrnn_persistent_kernel<<<NBLK, NTHR, 0, stream>>>(Ux, V_w, Ua_w, Va_w, v,
                                                      mem, uam, q, st, e,
                                                      ctpart, bar);
}